// BAttnDecoderLSTM_42039139893566
// MI455X (gfx1250) — compile-verified
//
#include <hip/hip_runtime.h>
#include <hip/hip_bf16.h>

// ---------------------------------------------------------------------------
// Problem constants (from reference setup_inputs)
// ---------------------------------------------------------------------------
constexpr int B   = 1024;   // batch
constexpr int NV  = 36;     // panoramic views
constexpr int F   = 2052;   // feature dim
constexpr int H   = 512;    // hidden
constexpr int E   = 64;     // action embed
constexpr int L   = 80;     // ctx length
constexpr int CAND= 16;     // candidates
constexpr int AF  = 4;      // angle feat
constexpr int K1  = E + F;      // 2116 : LSTM input width (a | attn_feat)
constexpr int G4  = 4 * H;      // 2048 : gates width
constexpr int D2  = 2 * H;      // 1024 : ctx dim
constexpr int KC  = 3 * H;      // 1536 : [wctx | h1] width

typedef float v2f __attribute__((ext_vector_type(2)));
typedef float v8f __attribute__((ext_vector_type(8)));

// ---------------------------------------------------------------------------
// Generic WMMA f32 GEMM:  C[M,N] (+)= A[M,K] @ W[N,K]^T  (+bias1+bias2)(tanh)
// A rows and W rows are both contiguous along K.
//
// Wave tile: 32(M) x 64(N) via 2x4 V_WMMA_F32_16X16X4_F32 fragments
//   -> per 4-K step: 2 A-frag + 4 B-frag loads (1.5 KB/wave) for 8 WMMAs
//      (16384 FLOP) = 10.7 FLOP/B of cache traffic.
// Block: 256 threads = 8 waves as 2(M) x 4(N) -> 64(M) x 256(N) per block.
//
// Fragment layouts per CDNA5 ISA 7.12.2 (32-bit, wave32):
//   A 16x4 : lane l (m = l&15, h = l>>4), VGPR j holds K = k0 + 2*h + j
//   B 4x16 : lane l (n = l&15, h = l>>4), VGPR j holds K = k0 + 2*h + j
//   C 16x16: VGPR p, lane l -> (m = p + 8*(l>>4), n = l&15)
// So both A and B fragments are a single float2 load per lane.
// ---------------------------------------------------------------------------
__global__ __launch_bounds__(256)
void wmma_gemm_nt(const float* __restrict__ A, const float* __restrict__ W,
                  float* __restrict__ Cout, int M, int N, int K,
                  const float* __restrict__ bias1, const float* __restrict__ bias2,
                  int accumulate, int act_tanh) {
    const int tid  = threadIdx.x;
    const int wave = tid >> 5;
    const int lane = tid & 31;
    const int r    = lane & 15;
    const int hh   = lane >> 4;
    const int wm   = wave >> 2;      // 0..1
    const int wn   = wave & 3;       // 0..3

    const int m0 = blockIdx.y * 64 + wm * 32;       // M always multiple of 64 here
    const int n0 = blockIdx.x * 256 + wn * 64;

    // K-contiguous row pointers (clamp N-edge rows; results masked at store)
    const float* __restrict__ arow0 = A + (size_t)(m0 + r) * K;
    const float* __restrict__ arow1 = A + (size_t)(m0 + 16 + r) * K;
    const float* wrow[4];
    #pragma unroll
    for (int j = 0; j < 4; ++j) {
        int nr = n0 + j * 16 + r;
        if (nr > N - 1) nr = N - 1;
        wrow[j] = W + (size_t)nr * K;
    }

    const int koff = 2 * hh;

    v8f c0[4] = {{}, {}, {}, {}};   // M sub-tile 0, N sub-tiles 0..3
    v8f c1[4] = {{}, {}, {}, {}};   // M sub-tile 1

    #pragma unroll 2
    for (int k0 = 0; k0 < K; k0 += 4) {
        v2f a0 = *(const v2f*)(arow0 + k0 + koff);
        v2f a1 = *(const v2f*)(arow1 + k0 + koff);
        v2f b0 = *(const v2f*)(wrow[0] + k0 + koff);
        v2f b1 = *(const v2f*)(wrow[1] + k0 + koff);
        v2f b2 = *(const v2f*)(wrow[2] + k0 + koff);
        v2f b3 = *(const v2f*)(wrow[3] + k0 + koff);
        c0[0] = __builtin_amdgcn_wmma_f32_16x16x4_f32(false, a0, false, b0, (short)0, c0[0], false, false);
        c0[1] = __builtin_amdgcn_wmma_f32_16x16x4_f32(false, a0, false, b1, (short)0, c0[1], false, false);
        c0[2] = __builtin_amdgcn_wmma_f32_16x16x4_f32(false, a0, false, b2, (short)0, c0[2], false, false);
        c0[3] = __builtin_amdgcn_wmma_f32_16x16x4_f32(false, a0, false, b3, (short)0, c0[3], false, false);
        c1[0] = __builtin_amdgcn_wmma_f32_16x16x4_f32(false, a1, false, b0, (short)0, c1[0], false, false);
        c1[1] = __builtin_amdgcn_wmma_f32_16x16x4_f32(false, a1, false, b1, (short)0, c1[1], false, false);
        c1[2] = __builtin_amdgcn_wmma_f32_16x16x4_f32(false, a1, false, b2, (short)0, c1[2], false, false);
        c1[3] = __builtin_amdgcn_wmma_f32_16x16x4_f32(false, a1, false, b3, (short)0, c1[3], false, false);
    }

    // ---- epilogue (divergence OK here; all WMMAs already issued) ----
    auto emit = [&](int row, int col, float v) {
        if (col < N) {
            size_t o = (size_t)row * N + col;
            float x = v;
            if (accumulate) x += Cout[o];
            if (bias1)      x += bias1[col];
            if (bias2)      x += bias2[col];
            if (act_tanh)   x = tanhf(x);
            Cout[o] = x;
        }
    };
    #pragma unroll
    for (int j = 0; j < 4; ++j) {
        int col = n0 + j * 16 + (lane & 15);
        #pragma unroll
        for (int p = 0; p < 8; ++p) {
            int row0 = m0 + p + 8 * hh;
            emit(row0,      col, c0[j][p]);
            emit(row0 + 16, col, c1[j][p]);
        }
    }
}

// ---------------------------------------------------------------------------
// Block reduction helper: 256 threads = 8 waves (wave32)
// ---------------------------------------------------------------------------
__device__ __forceinline__ float block_reduce_sum(float v, float* smem8) {
    #pragma unroll
    for (int off = 16; off > 0; off >>= 1)
        v += __shfl_down(v, off, 32);
    int tid = threadIdx.x;
    if ((tid & 31) == 0) smem8[tid >> 5] = v;
    __syncthreads();
    if (tid == 0) {
        float r = smem8[0];
        #pragma unroll
        for (int i = 1; i < 8; ++i) r += smem8[i];
        smem8[0] = r;
    }
    __syncthreads();
    float out = smem8[0];
    __syncthreads();
    return out;
}

// ---------------------------------------------------------------------------
// a = tanh(action @ W_emb^T + b_emb)  -> x_cat[:, 0:E]
// ---------------------------------------------------------------------------
__global__ __launch_bounds__(256)
void action_embed_kernel(const float* __restrict__ action,
                         const float* __restrict__ W_emb,
                         const float* __restrict__ b_emb,
                         float* __restrict__ x_cat) {
    int idx = blockIdx.x * 256 + threadIdx.x;
    if (idx >= B * E) return;
    int b = idx >> 6, e = idx & 63;
    float s = b_emb[e];
    #pragma unroll
    for (int k = 0; k < AF; ++k)
        s += action[b * AF + k] * W_emb[e * AF + k];
    x_cat[(size_t)b * K1 + e] = tanhf(s);
}

// ---------------------------------------------------------------------------
// Shift-softmax attention over panoramic features (one block per batch row).
// Streams `feature` twice (row working set 295 KB -> 2nd pass hits L2).
// Writes attn_feat into x_cat[:, E:E+F].
// ---------------------------------------------------------------------------
__global__ __launch_bounds__(256)
void shift_attn_kernel(const float* __restrict__ feature,
                       const float* __restrict__ tgt,      // [B,F] = prev_h1 @ Win_f^T
                       const float* __restrict__ prev_h1,  // [B,H]
                       const float* __restrict__ Wsh,      // [3,H]
                       const float* __restrict__ bsh,      // [3]
                       float* __restrict__ x_cat) {
    const int b   = blockIdx.x;
    const int tid = threadIdx.x;
    __shared__ float red[8];
    __shared__ float sc[NV];
    __shared__ float at[NV];
    __shared__ float kl[3];

    const float* frow = feature + (size_t)b * NV * F;
    const float* trow = tgt + (size_t)b * F;

    // raw attention scores over 36 views
    for (int s = 0; s < NV; ++s) {
        float p = 0.f;
        for (int f = tid; f < F; f += 256)
            p += frow[(size_t)s * F + f] * trow[f];
        float tot = block_reduce_sum(p, red);
        if (tid == 0) sc[s] = tot;
    }
    // shift-kernel logits: prev_h1 @ Wsh^T + bsh
    const float* hrow = prev_h1 + (size_t)b * H;
    for (int j = 0; j < 3; ++j) {
        float p = 0.f;
        for (int f = tid; f < H; f += 256)
            p += hrow[f] * Wsh[j * H + f];
        float tot = block_reduce_sum(p, red);
        if (tid == 0) kl[j] = tot + bsh[j];
    }
    __syncthreads();

    if (tid == 0) {
        // softmax over 36 scores
        float m = -INFINITY;
        for (int s = 0; s < NV; ++s) m = fmaxf(m, sc[s]);
        float sum = 0.f;
        for (int s = 0; s < NV; ++s) { float e = expf(sc[s] - m); sc[s] = e; sum += e; }
        float inv = 1.f / sum;
        for (int s = 0; s < NV; ++s) sc[s] *= inv;
        // softmax over 3 kernel taps
        float km = fmaxf(kl[0], fmaxf(kl[1], kl[2]));
        float k0 = expf(kl[0] - km), k1 = expf(kl[1] - km), k2 = expf(kl[2] - km);
        float kinv = 1.f / (k0 + k1 + k2);
        k0 *= kinv; k1 *= kinv; k2 *= kinv;
        // grouped circular shift-conv: 3 groups of 12, kernel size 3, pad 1
        const int W = NV / 3;
        for (int g = 0; g < 3; ++g)
            for (int w = 0; w < W; ++w) {
                float acc = k0 * sc[g * W + (w + W - 1) % W]
                          + k1 * sc[g * W + w]
                          + k2 * sc[g * W + (w + 1) % W];
                at[g * W + w] = acc;
            }
    }
    __syncthreads();

    // attn_feat = sum_s at[s] * feature[b,s,:]  -> x_cat[:, E:]
    float* xrow = x_cat + (size_t)b * K1 + E;
    for (int f = tid; f < F; f += 256) {
        float acc = 0.f;
        #pragma unroll 4
        for (int s = 0; s < NV; ++s)
            acc += at[s] * frow[(size_t)s * F + f];
        xrow[f] = acc;
    }
}

// ---------------------------------------------------------------------------
// LSTM elementwise: gates [B,4H] (i,f,g,o) -> c_1, h_1 (into d_out)
// ---------------------------------------------------------------------------
__device__ __forceinline__ float sigmoidf(float x) { return 1.f / (1.f + expf(-x)); }

__global__ __launch_bounds__(256)
void lstm_cell_kernel(const float* __restrict__ gates,
                      const float* __restrict__ c_0,
                      float* __restrict__ h_1,
                      float* __restrict__ c_1) {
    int idx = blockIdx.x * 256 + threadIdx.x;
    if (idx >= B * H) return;
    int b = idx / H, n = idx % H;
    const float* g = gates + (size_t)b * G4;
    float gi = g[n], gf = g[H + n], gg = g[2 * H + n], go = g[3 * H + n];
    float c = sigmoidf(gf) * c_0[idx] + sigmoidf(gi) * tanhf(gg);
    c_1[idx] = c;
    h_1[idx] = sigmoidf(go) * tanhf(c);
}

// ---------------------------------------------------------------------------
// Masked ctx attention (one block per batch row). Streams ctx twice.
// Writes [wctx | h_1] into xc [B, 3H].
// ---------------------------------------------------------------------------
__global__ __launch_bounds__(256)
void ctx_attn_kernel(const float* __restrict__ ctx,      // [B,L,2H]
                     const int*   __restrict__ ctx_mask, // [B,L]
                     const float* __restrict__ tgt_c,    // [B,2H]
                     const float* __restrict__ h_1,      // [B,H]
                     float* __restrict__ xc) {           // [B,3H]
    const int b   = blockIdx.x;
    const int tid = threadIdx.x;
    __shared__ float red[8];
    __shared__ float sl[L];

    const float* crow = ctx + (size_t)b * L * D2;
    const float* trow = tgt_c + (size_t)b * D2;
    const int*   mrow = ctx_mask + (size_t)b * L;

    for (int l = 0; l < L; ++l) {
        float p = 0.f;
        for (int d = tid; d < D2; d += 256)
            p += crow[(size_t)l * D2 + d] * trow[d];
        float tot = block_reduce_sum(p, red);
        if (tid == 0) sl[l] = (mrow[l] != 0) ? -INFINITY : tot;
    }
    __syncthreads();
    if (tid == 0) {
        float m = -INFINITY;
        for (int l = 0; l < L; ++l) m = fmaxf(m, sl[l]);
        float sum = 0.f;
        for (int l = 0; l < L; ++l) { float e = expf(sl[l] - m); sl[l] = e; sum += e; }
        float inv = 1.f / sum;
        for (int l = 0; l < L; ++l) sl[l] *= inv;
    }
    __syncthreads();

    float* xrow = xc + (size_t)b * KC;
    for (int d = tid; d < D2; d += 256) {
        float acc = 0.f;
        #pragma unroll 4
        for (int l = 0; l < L; ++l)
            acc += sl[l] * crow[(size_t)l * D2 + d];
        xrow[d] = acc;
    }
    const float* hrow = h_1 + (size_t)b * H;
    for (int d = tid; d < H; d += 256)
        xrow[D2 + d] = hrow[d];
}

// ---------------------------------------------------------------------------
// Candidate logits (one block per batch row): 16 dots of length F.
// ---------------------------------------------------------------------------
__global__ __launch_bounds__(256)
void cand_logit_kernel(const float* __restrict__ cand_feat, // [B,C,F]
                       const float* __restrict__ tgt_k,     // [B,F]
                       float* __restrict__ logit) {         // [B,C]
    const int b   = blockIdx.x;
    const int tid = threadIdx.x;
    __shared__ float red[8];
    const float* crow = cand_feat + (size_t)b * CAND * F;
    const float* trow = tgt_k + (size_t)b * F;
    for (int c = 0; c < CAND; ++c) {
        float p = 0.f;
        for (int f = tid; f < F; f += 256)
            p += crow[(size_t)c * F + f] * trow[f];
        float tot = block_reduce_sum(p, red);
        if (tid == 0) logit[(size_t)b * CAND + c] = tot;
    }
}

// ---------------------------------------------------------------------------
// Launch
// ---------------------------------------------------------------------------
extern "C" void kernel_launch(void* const* d_in, const int* in_sizes, int n_in,
                              void* d_out, int out_size, void* d_ws, size_t ws_size,
                              hipStream_t stream) {
    const float* action   = (const float*)d_in[0];
    const float* feature  = (const float*)d_in[1];
    const float* cand     = (const float*)d_in[2];
    /* d_in[3] = h_0 (unused by reference) */
    const float* prev_h1  = (const float*)d_in[4];
    const float* c_0      = (const float*)d_in[5];
    const float* ctx      = (const float*)d_in[6];
    const int*   ctx_mask = (const int*)d_in[7];
    const float* W_emb    = (const float*)d_in[8];
    const float* b_emb    = (const float*)d_in[9];
    const float* Win_f    = (const float*)d_in[10];
    const float* Wsh      = (const float*)d_in[11];
    const float* bsh      = (const float*)d_in[12];
    const float* Wih      = (const float*)d_in[13];
    const float* Whh      = (const float*)d_in[14];
    const float* bih      = (const float*)d_in[15];
    const float* bhh      = (const float*)d_in[16];
    const float* Win_c    = (const float*)d_in[17];
    const float* Wout_c   = (const float*)d_in[18];
    const float* Win_k    = (const float*)d_in[19];

    // workspace layout (floats); buffers reused across phases
    float* x_cat = (float*)d_ws;                  // [B,K1]  (a | attn_feat); later reused for tgt_c [B,2H]
    float* buf1  = x_cat + (size_t)B * K1;        // [B,F]   tgt_f; later tgt_k
    float* buf2  = buf1 + (size_t)B * F;          // [B,4H]  gates; later xc [B,3H]

    // output layout: h_1 | c_1 | logit | h_tilde
    float* h_1     = (float*)d_out;
    float* c_1     = h_1 + (size_t)B * H;
    float* logit   = c_1 + (size_t)B * H;
    float* h_tilde = logit + (size_t)B * CAND;

    const dim3 blk(256);

    // 1) action embedding -> x_cat[:, :E]
    action_embed_kernel<<<(B * E + 255) / 256, blk, 0, stream>>>(action, W_emb, b_emb, x_cat);

    // 2) tgt_f = prev_h1 @ Win_f^T   [B,F]
    wmma_gemm_nt<<<dim3((F + 255) / 256, B / 64), blk, 0, stream>>>(
        prev_h1, Win_f, buf1, B, F, H, nullptr, nullptr, 0, 0);

    // 3) shift attention -> x_cat[:, E:]
    shift_attn_kernel<<<B, blk, 0, stream>>>(feature, buf1, prev_h1, Wsh, bsh, x_cat);

    // 4) gates = x_cat @ Wih^T                       [B,4H], K=2116
    wmma_gemm_nt<<<dim3(G4 / 256, B / 64), blk, 0, stream>>>(
        x_cat, Wih, buf2, B, G4, K1, nullptr, nullptr, 0, 0);
    // 5) gates += prev_h1 @ Whh^T + bih + bhh        K=512
    wmma_gemm_nt<<<dim3(G4 / 256, B / 64), blk, 0, stream>>>(
        prev_h1, Whh, buf2, B, G4, H, bih, bhh, 1, 0);

    // 6) LSTM cell -> h_1, c_1 (d_out)
    lstm_cell_kernel<<<(B * H + 255) / 256, blk, 0, stream>>>(buf2, c_0, h_1, c_1);

    // 7) tgt_c = h_1 @ Win_c^T  [B,2H]  (reuse x_cat)
    wmma_gemm_nt<<<dim3(D2 / 256, B / 64), blk, 0, stream>>>(
        h_1, Win_c, x_cat, B, D2, H, nullptr, nullptr, 0, 0);

    // 8) masked ctx attention -> xc = [wctx | h_1]  (reuse buf2)
    ctx_attn_kernel<<<B, blk, 0, stream>>>(ctx, ctx_mask, x_cat, h_1, buf2);

    // 9) h_tilde = tanh(xc @ Wout_c^T)  [B,H] -> d_out
    wmma_gemm_nt<<<dim3((H + 255) / 256, B / 64), blk, 0, stream>>>(
        buf2, Wout_c, h_tilde, B, H, KC, nullptr, nullptr, 0, 1);

    // 10) tgt_k = h_tilde @ Win_k^T  [B,F]  (reuse buf1)
    wmma_gemm_nt<<<dim3((F + 255) / 256, B / 64), blk, 0, stream>>>(
        h_tilde, Win_k, buf1, B, F, H, nullptr, nullptr, 0, 0);

    // 11) candidate logits -> d_out
    cand_logit_kernel<<<B, blk, 0, stream>>>(cand, buf1, logit);
}